// LossWithEuler_17970143166491
// MI455X (gfx1250) — compile-verified
//
#include <hip/hip_runtime.h>
#include <hip/hip_bf16.h>

// Problem shape (fixed by the reference).
#define LWE_D       136
#define LWE_B       131072LL
#define LWE_DCHUNKS 4            // 136 / 4 = 34 rows per chunk
#define LWE_DROWS   34
#define LWE_CBLKS   128          // column blocks: 128 * 256 threads * 4 cols = 131072
#define LWE_NPART   (LWE_CBLKS * LWE_DCHUNKS)   // 512 block partials

typedef float v2f __attribute__((ext_vector_type(2)));
typedef float v4f __attribute__((ext_vector_type(4)));
typedef float v8f __attribute__((ext_vector_type(8)));
typedef int   v4i __attribute__((ext_vector_type(4)));

// Non-temporal 128-bit load: the inp/label stream is single-use (143 MB),
// keep it out of the way of the reused ea/attribute data in cache.
__device__ __forceinline__ v4f ld_nt_v4f(const float* p) {
    return __builtin_nontemporal_load((const v4f*)p);
}

// Stage 1: each thread owns 4 consecutive columns; compute weight[b] once,
// accumulate sum_d (inp-lab)^2 over a 34-row D-chunk, fold weight in at the
// end, reduce the block, emit one partial per block. Fully deterministic.
__global__ void __launch_bounds__(256)
lwe_partial_kernel(const float* __restrict__ inp,
                   const float* __restrict__ lab,
                   const float* __restrict__ ea,
                   const int*   __restrict__ attr,
                   const float* __restrict__ an,
                   float*       __restrict__ ws) {
    const int  g  = blockIdx.x * 256 + threadIdx.x;     // column-group id
    const long long b0 = (long long)g * 4;              // first of 4 columns

    // Inverse-frequency table (uniform across the wave -> scalar cached).
    const float a0 = an[0], a1 = an[1], a2 = an[2], a3 = an[3], a4 = an[4], a5 = an[5];
    const float tot = a0 + a1 + a2 + a3 + a4 + a5;
    const float inv[6] = { tot / a0, tot / a1, tot / a2, tot / a3, tot / a4, tot / a5 };

    // attr_w[b] = sum_j (attribute[j,b]==1) * inv_freq[j]
    v4f attrw = {0.f, 0.f, 0.f, 0.f};
#pragma unroll
    for (int j = 0; j < 6; ++j) {
        v4i aj = *(const v4i*)(attr + (long long)j * LWE_B + b0);
        attrw.x += (aj.x == 1) ? inv[j] : 0.f;
        attrw.y += (aj.y == 1) ? inv[j] : 0.f;
        attrw.z += (aj.z == 1) ? inv[j] : 0.f;
        attrw.w += (aj.w == 1) ? inv[j] : 0.f;
    }

    // angle_w[b] = sum_j (1 - cos(ea[j,b])) = 3 - sum_j cos(ea[j,b])
    v4f angw = {3.f, 3.f, 3.f, 3.f};
#pragma unroll
    for (int j = 0; j < 3; ++j) {
        v4f e = *(const v4f*)(ea + (long long)j * LWE_B + b0);
        angw.x -= cosf(e.x);
        angw.y -= cosf(e.y);
        angw.z -= cosf(e.z);
        angw.w -= cosf(e.w);
    }
    const v4f w = attrw * angw;

    // Streaming accumulation of squared diffs over this D-chunk.
    const long long base = (long long)(blockIdx.y * LWE_DROWS) * LWE_B + b0;
    const float* pi = inp + base;
    const float* pl = lab + base;
    v4f acc = {0.f, 0.f, 0.f, 0.f};
#pragma unroll 4
    for (int d = 0; d < LWE_DROWS; ++d) {
        v4f x  = ld_nt_v4f(pi);
        v4f y  = ld_nt_v4f(pl);
        v4f df = x - y;
        acc.x = fmaf(df.x, df.x, acc.x);
        acc.y = fmaf(df.y, df.y, acc.y);
        acc.z = fmaf(df.z, df.z, acc.z);
        acc.w = fmaf(df.w, df.w, acc.w);
        pi += LWE_B;
        pl += LWE_B;
    }
    float p = w.x * acc.x + w.y * acc.y + w.z * acc.z + w.w * acc.w;

    // wave32 shuffle reduction, then 8-wave LDS reduction.
#pragma unroll
    for (int off = 16; off > 0; off >>= 1) p += __shfl_down(p, off, 32);

    __shared__ float sm[8];
    if ((threadIdx.x & 31) == 0) sm[threadIdx.x >> 5] = p;
    __syncthreads();
    if (threadIdx.x == 0) {
        float t = 0.f;
#pragma unroll
        for (int i = 0; i < 8; ++i) t += sm[i];
        ws[blockIdx.y * gridDim.x + blockIdx.x] = t;
    }
}

// Stage 2: single wave sums the 512 partials. The final 32-lane cross-lane
// reduction is done with V_WMMA_F32_16X16X4_F32: A holds the lane partials
// (VGPR0 -> K=0 for lanes 0-15, K=2 for lanes 16-31; VGPR1 = 0), B = ones,
// so D[m,n] = s_m + s_{m+16}; summing column 0 (lane0 + lane16 D registers)
// yields the exact f32 total.
__global__ void __launch_bounds__(32)
lwe_final_kernel(const float* __restrict__ ws, float* __restrict__ out) {
    const int lane = threadIdx.x;
    float s = 0.f;
#pragma unroll
    for (int i = 0; i < LWE_NPART / 32; ++i) s += ws[lane + 32 * i];

#if __has_builtin(__builtin_amdgcn_wmma_f32_16x16x4_f32)
    v2f a;   a.x = s;   a.y = 0.f;
    v2f one; one.x = 1.f; one.y = 1.f;
    v8f c = {0.f, 0.f, 0.f, 0.f, 0.f, 0.f, 0.f, 0.f};
    v8f dm = __builtin_amdgcn_wmma_f32_16x16x4_f32(
        /*neg_a=*/false, a, /*neg_b=*/false, one,
        /*c_mod=*/(short)0, c, /*reuse_a=*/false, /*reuse_b=*/false);
    // lane0 holds D[0..7,0]; lane16 holds D[8..15,0].
    float cs = dm[0] + dm[1] + dm[2] + dm[3] + dm[4] + dm[5] + dm[6] + dm[7];
    float total = cs + __shfl_down(cs, 16, 32);
#else
    float total = s;
#pragma unroll
    for (int off = 16; off > 0; off >>= 1) total += __shfl_down(total, off, 32);
#endif

    if (lane == 0)
        out[0] = total * (1.0f / (float)((long long)LWE_D * LWE_B));
}

extern "C" void kernel_launch(void* const* d_in, const int* in_sizes, int n_in,
                              void* d_out, int out_size, void* d_ws, size_t ws_size,
                              hipStream_t stream) {
    const float* inp  = (const float*)d_in[0];
    const float* lab  = (const float*)d_in[1];
    const float* ea   = (const float*)d_in[2];
    const int*   attr = (const int*)  d_in[3];
    const float* an   = (const float*)d_in[4];
    float* out = (float*)d_out;
    float* ws  = (float*)d_ws;   // 512 floats of scratch, fully rewritten each call

    dim3 grid(LWE_CBLKS, LWE_DCHUNKS);
    lwe_partial_kernel<<<grid, 256, 0, stream>>>(inp, lab, ea, attr, an, ws);
    lwe_final_kernel<<<1, 32, 0, stream>>>(ws, out);
}